// LSSViewTransformer_72945724555285
// MI455X (gfx1250) — compile-verified
//
#include <hip/hip_runtime.h>

typedef __attribute__((ext_vector_type(2))) float v2f;
typedef __attribute__((ext_vector_type(4))) float v4f;
typedef __attribute__((ext_vector_type(8))) float v8f;

#define NCAM  6
#define NCAMP 8                  // camera dim padded for K=8 (2x WMMA K=4)
#define NCH   128
#define ND    40
#define NDP   48                 // depth dim padded to 3 N-tiles of 16
#define HH    32
#define WW    88
#define NPIX  (HH * WW)          // 2816
#define BEVH  125
#define BEVW  125
#define NBEV  (BEVH * BEVW)      // 15625

// ---------------------------------------------------------------- zero output
__global__ void zero_out_kernel(float* __restrict__ out, int n) {
    int i = blockIdx.x * blockDim.x + threadIdx.x;
    if (i < n) out[i] = 0.0f;
}

// ---------------------------------------------- static voxel index (padded)
// idxpad[pix][dcol], dcol in [0,48): -1 for invalid or dcol >= 40.
// Faithful constants: max1 = 87*41 = 3567, max2 = max(50, 31*41=1271) = 1271.
__global__ void idx_kernel(int* __restrict__ idxpad) {
    int i = blockIdx.x * blockDim.x + threadIdx.x;   // pix*NDP + dcol
    if (i >= NPIX * NDP) return;
    int dcol = i % NDP;
    int pix  = i / NDP;
    int w = pix % WW;
    int h = pix / WW;
    int res = -1;
    if (dcol < ND) {
        float depth = (float)(dcol + 2);
        float xd = (float)w * depth;
        float yd = (float)h * depth;
        float gx = xd / 3567.0f * 100.0f - 50.0f;
        float gy = yd / 1271.0f * 100.0f - 50.0f;
        int ix = (int)((gx + 50.0f) / 0.8f);
        int iy = (int)((gy + 50.0f) / 0.8f);
        int iz = (int)((depth + 10.0f) / 20.0f);
        if (ix >= 0 && ix < BEVW && iy >= 0 && iy < BEVH && iz >= 0)
            res = iy * BEVW + ix;
    }
    idxpad[i] = res;
}

// ------------------------------------------- repack img_feat: apad[pix][c][cam8]
// Zero-padded cameras 6,7 give the WMMA K=8 padding for free.
__global__ void repack_kernel(const float* __restrict__ img,
                              float* __restrict__ apad) {
    int i = blockIdx.x * blockDim.x + threadIdx.x;   // pix*NCH + c
    if (i >= NPIX * NCH) return;
    int c   = i % NCH;
    int pix = i / NCH;
    float v0 = img[((size_t)(0 * NCH + c)) * NPIX + pix];
    float v1 = img[((size_t)(1 * NCH + c)) * NPIX + pix];
    float v2 = img[((size_t)(2 * NCH + c)) * NPIX + pix];
    float v3 = img[((size_t)(3 * NCH + c)) * NPIX + pix];
    float v4 = img[((size_t)(4 * NCH + c)) * NPIX + pix];
    float v5 = img[((size_t)(5 * NCH + c)) * NPIX + pix];
    v4f lo = {v0, v1, v2, v3};
    v4f hi = {v4, v5, 0.0f, 0.0f};
    *(v4f*)(apad + (size_t)i * NCAMP)     = lo;       // 32B-aligned B128 store
    *(v4f*)(apad + (size_t)i * NCAMP + 4) = hi;
}

// --------------------------------------------------- softmax over depth bins
// One wave per pixel. Output: ppad[pix][d48][cam8], pad region zeroed first
// (same-wave stores are ordered, so zero-then-overwrite is safe).
__global__ void softmax_kernel(const float* __restrict__ logits,
                               float* __restrict__ probs) {
    int lane = threadIdx.x & 31;
    int wv   = (blockIdx.x * blockDim.x + threadIdx.x) >> 5;
    if (wv >= NPIX) return;
    float* blk = probs + (size_t)wv * (NDP * NCAMP);  // 384 floats
    #pragma unroll
    for (int j = 0; j < NDP * NCAMP / 32; ++j)        // 12 stores/lane
        blk[j * 32 + lane] = 0.0f;
    for (int n = 0; n < NCAM; ++n) {
        const float* base = logits + (size_t)(n * ND) * NPIX + wv;
        float x0 = base[(size_t)lane * NPIX];
        float x1 = (lane < 8) ? base[(size_t)(32 + lane) * NPIX] : -3.4e38f;
        float m = fmaxf(x0, x1);
        #pragma unroll
        for (int off = 16; off; off >>= 1) m = fmaxf(m, __shfl_xor(m, off, 32));
        float e0 = __expf(x0 - m);
        float e1 = (lane < 8) ? __expf(x1 - m) : 0.0f;
        float s = e0 + e1;
        #pragma unroll
        for (int off = 16; off; off >>= 1) s += __shfl_xor(s, off, 32);
        float inv = 1.0f / s;
        blk[(size_t)lane * NCAMP + n] = e0 * inv;
        if (lane < 8) blk[(size_t)(32 + lane) * NCAMP + n] = e1 * inv;
    }
}

// ----------------------------------------------------- WMMA contract + splat
// One wave per (pixel, 16-channel tile):
//   Q[128x40] = A[128xK8(cams, zero-padded)] @ P[K8x48]  via v_wmma_f32_16x16x4_f32
// All loads unconditional (padding lives in the buffers); EXEC stays all-1s
// through the WMMA region.  Scatter via global_atomic_add_f32.
__global__ void splat_kernel(const float* __restrict__ apad,
                             const float* __restrict__ ppad,
                             const int* __restrict__ idxpad,
                             float* __restrict__ out) {
    int lane = threadIdx.x & 31;
    int wv   = blockIdx.x * (blockDim.x >> 5) + (threadIdx.x >> 5);
    int pix  = wv >> 3;
    int ct   = wv & 7;                 // channel tile 0..7
    int m    = lane & 15;
    int half = lane >> 4;
    int k0   = half * 2;               // A 16x4 layout: lanes 0-15 K=0,1; 16-31 K=2,3
    int c    = ct * 16 + m;

    const float* ap = apad + ((size_t)pix * NCH + c) * NCAMP;
    v2f a0 = *(const v2f*)(ap + k0);       // K = {0,1} / {2,3}
    v2f a1 = *(const v2f*)(ap + k0 + 4);   // K = {4,5} / {6,7}(zeros)

    // Hoist all BEV index loads so they fly with the front load clause and
    // their latency hides under the WMMA chain (no mid-scatter stalls).
    const int* ip = idxpad + (size_t)pix * NDP;
    int bidx0 = ip[m];
    int bidx1 = ip[16 + m];
    int bidx2 = ip[32 + m];               // -1 encodes invalid OR dcol >= 40

    const float* pp = ppad + (size_t)pix * (NDP * NCAMP);
    v8f acc[3];
    acc[0] = (v8f){};  acc[1] = (v8f){};  acc[2] = (v8f){};

    #pragma unroll
    for (int t = 0; t < 3; ++t) {
        const float* pt = pp + (size_t)(t * 16 + m) * NCAMP + k0;
        v2f b0 = *(const v2f*)(pt);        // P[k0..k0+1][dcol]
        v2f b1 = *(const v2f*)(pt + 4);    // P[k0+4..k0+5][dcol] (pads are zero)
        acc[t] = __builtin_amdgcn_wmma_f32_16x16x4_f32(
            false, a0, false, b0, (short)0, acc[t], false, false);
        acc[t] = __builtin_amdgcn_wmma_f32_16x16x4_f32(
            false, a1, false, b1, (short)0, acc[t], false, false);
    }

    // Scatter: D-matrix layout VGPR r -> M = r + 8*half, N = lane&15
    int bidx[3] = {bidx0, bidx1, bidx2};
    #pragma unroll
    for (int t = 0; t < 3; ++t) {
        if (bidx[t] >= 0) {
            #pragma unroll
            for (int r = 0; r < 8; ++r) {
                int cc = ct * 16 + r + half * 8;
                atomicAdd(out + (size_t)cc * NBEV + bidx[t], acc[t][r]);
            }
        }
    }
}

// ---------------------------------------------------------------------------
extern "C" void kernel_launch(void* const* d_in, const int* in_sizes, int n_in,
                              void* d_out, int out_size, void* d_ws, size_t ws_size,
                              hipStream_t stream) {
    const float* img    = (const float*)d_in[0];   // (1,6,128,32,88)
    const float* logits = (const float*)d_in[1];   // (1,6,40,32,88)
    // d_in[2..4] (rots, trans, intrins) are unused by the reference math.
    float* out = (float*)d_out;                    // (1,128,125,125)

    // Workspace layout (all 32B-aligned):
    //   apad  : NPIX*NCH*NCAMP  f32 = 11.53 MB
    //   ppad  : NPIX*NDP*NCAMP  f32 =  4.33 MB
    //   idxpad: NPIX*NDP        i32 =  0.54 MB
    float* apad   = (float*)d_ws;
    float* ppad   = apad + (size_t)NPIX * NCH * NCAMP;
    int*   idxpad = (int*)(ppad + (size_t)NPIX * NDP * NCAMP);

    zero_out_kernel<<<(out_size + 255) / 256, 256, 0, stream>>>(out, out_size);
    idx_kernel<<<(NPIX * NDP + 255) / 256, 256, 0, stream>>>(idxpad);
    repack_kernel<<<(NPIX * NCH + 255) / 256, 256, 0, stream>>>(img, apad);
    softmax_kernel<<<NPIX / 4, 128, 0, stream>>>(logits, ppad);             // 704 blocks
    splat_kernel<<<(NPIX * 8) / 4, 128, 0, stream>>>(apad, ppad, idxpad, out); // 5632 blocks
}